// CapsLayer_52656299049586
// MI455X (gfx1250) — compile-verified
//
#include <hip/hip_runtime.h>
#include <hip/hip_bf16.h>

typedef __attribute__((ext_vector_type(2))) float v2f;
typedef __attribute__((ext_vector_type(4))) float v4f;
typedef __attribute__((ext_vector_type(8))) float v8f;

#define B_DIM 32
#define R_DIM 512
#define C_DIM 32
#define O_DIM 64
#define I_DIM 128
#define N_DIM (C_DIM * O_DIM)   // 2048
#define XS_STRIDE 132           // 128 + 4 pad: keeps 16B alignment, rotates banks by 4/row

// -------------------------------------------------------------------------
// K1: u_hat[b,r,c,o] = sum_i W[r,c,o,i] * x[b,r,i]
// Per r: GEMM M=32(B), N=2048(C*O), K=128(I) using V_WMMA_F32_16X16X4_F32.
// grid = (16 n-groups, 512 r), block = 256 (8 waves), wave -> 32x16 tile.
// -------------------------------------------------------------------------
__global__ __launch_bounds__(256)
void uhat_gemm(const float* __restrict__ x, const float* __restrict__ W,
               float* __restrict__ uhat) {
    __shared__ float xs[B_DIM * XS_STRIDE];

    const int r = blockIdx.y;

    // Stage x[:, r, :] (32 rows x 128 floats) into LDS, coalesced float4.
    for (int f = threadIdx.x; f < B_DIM * (I_DIM / 4); f += 256) {
        const int bb = f >> 5;            // batch row
        const int i4 = (f & 31) * 4;      // float4 column
        v4f val = *(const v4f*)(x + ((size_t)bb * R_DIM + r) * I_DIM + i4);
        *(v4f*)(&xs[bb * XS_STRIDE + i4]) = val;
    }
    __syncthreads();

    const int lane = threadIdx.x & 31;
    const int wv   = threadIdx.x >> 5;                 // wave id 0..7
    const int n0   = blockIdx.x * 128 + wv * 16;       // this wave's N base
    const int lh   = lane & 15;                        // lane within half
    const int half = lane >> 4;                        // 0 or 1
    const int acol = 2 * half;                         // K sub-offset per ISA layout

    // B operand: lane holds W[r, n0+lh, k0 + acol + {0,1}] per K-step.
    const float* wrow = W + ((size_t)r * N_DIM + n0 + lh) * I_DIM + acol;
    // A operand rows in LDS (M striped in lanes 0..15; K split across halves).
    const float* a0p = xs + lh * XS_STRIDE + acol;            // M = 0..15
    const float* a1p = xs + (16 + lh) * XS_STRIDE + acol;     // M = 16..31

    v8f acc0 = {};
    v8f acc1 = {};
#pragma unroll 4
    for (int k0 = 0; k0 < I_DIM; k0 += 4) {
        if ((k0 & 31) == 0) {
            // One prefetch per 128B cacheline of this lane's W row, pulled
            // into near (WGP) cache: locality 3 -> RT non-speculative near.
            __builtin_prefetch(wrow + k0 + 32, 0, 3);
        }
        v2f bvec = *(const v2f*)(wrow + k0);
        v2f av0  = *(const v2f*)(a0p + k0);
        v2f av1  = *(const v2f*)(a1p + k0);
        acc0 = __builtin_amdgcn_wmma_f32_16x16x4_f32(
            false, av0, false, bvec, (short)0, acc0, false, false);
        acc1 = __builtin_amdgcn_wmma_f32_16x16x4_f32(
            false, av1, false, bvec, (short)0, acc1, false, false);
    }

    // D layout: VGPR j, lanes 0-15 -> M=j, N=lane; lanes 16-31 -> M=8+j, N=lane-16.
    float* up = uhat + (size_t)r * N_DIM + n0 + lh;
    const int mbase = 8 * half;
    const size_t bstride = (size_t)R_DIM * N_DIM;
#pragma unroll
    for (int j = 0; j < 8; ++j) {
        up[(size_t)(mbase + j) * bstride]      = acc0[j];
        up[(size_t)(mbase + j + 16) * bstride] = acc1[j];
    }
}

// -------------------------------------------------------------------------
// K2: c_ij = softmax(b_ij, axis=C). One wave per (b,r); lane = c (wave32).
// -------------------------------------------------------------------------
__global__ __launch_bounds__(256)
void softmax_c(const float* __restrict__ b_ij, float* __restrict__ c_ij) {
    const int lane = threadIdx.x & 31;
    const int idx  = blockIdx.x * 8 + (threadIdx.x >> 5);   // (b*512 + r)
    const float bv = b_ij[(size_t)idx * C_DIM + lane];
    float m = bv;
    for (int off = 16; off; off >>= 1) m = fmaxf(m, __shfl_xor(m, off, 32));
    const float e = __expf(bv - m);
    float ssum = e;
    for (int off = 16; off; off >>= 1) ssum += __shfl_xor(ssum, off, 32);
    c_ij[(size_t)idx * C_DIM + lane] = e / ssum;
}

// -------------------------------------------------------------------------
// K3: s[b,c,o] = sum_r c_ij[b,r,c] * u_hat[b,r,c,o]. Block per (c,b);
// 256 threads = 4 r-lanes x 64 o (coalesced 256B reads of u_hat).
// -------------------------------------------------------------------------
__global__ __launch_bounds__(256)
void reduce_s(const float* __restrict__ c_ij, const float* __restrict__ uhat,
              float* __restrict__ s) {
    __shared__ float red[256];
    const int c  = blockIdx.x;
    const int b  = blockIdx.y;
    const int o  = threadIdx.x & 63;
    const int rl = threadIdx.x >> 6;

    float acc = 0.0f;
    for (int r = rl; r < R_DIM; r += 4) {
        const float cij = c_ij[((size_t)b * R_DIM + r) * C_DIM + c];
        const float u = uhat[((size_t)b * R_DIM + r) * N_DIM + c * O_DIM + o];
        acc = fmaf(cij, u, acc);
    }
    red[threadIdx.x] = acc;
    __syncthreads();
    if (rl == 0) {
        s[((size_t)b * C_DIM + c) * O_DIM + o] =
            red[o] + red[64 + o] + red[128 + o] + red[192 + o];
    }
}

// -------------------------------------------------------------------------
// K4: elementwise squash v = s^2 * s / ((1+s^2) * sqrt(s^2 + eps))
// -------------------------------------------------------------------------
__global__ __launch_bounds__(256)
void squash_v(const float* __restrict__ s, float* __restrict__ v) {
    const int i = blockIdx.x * 256 + threadIdx.x;
    const float sv = s[i];
    const float sq = sv * sv;
    v[i] = sq * sv / ((1.0f + sq) * sqrtf(sq + 1e-8f));
}

// -------------------------------------------------------------------------
// K5: b_ij[b,r,c] += sum_o u_hat[b,r,c,o] * v[b,c,o]. Block per (r,b);
// wave per c (4 c's per wave), float2 per lane over O=64, shfl reduce.
// -------------------------------------------------------------------------
__global__ __launch_bounds__(256)
void agree_update(const float* __restrict__ uhat, const float* __restrict__ v,
                  float* __restrict__ b_ij) {
    const int r = blockIdx.x;
    const int b = blockIdx.y;
    const int lane = threadIdx.x & 31;
    const int wv   = threadIdx.x >> 5;

    const float* up = uhat + ((size_t)b * R_DIM + r) * N_DIM;
    const float* vp = v + (size_t)b * N_DIM;

    for (int c = wv; c < C_DIM; c += 8) {
        v2f u2 = *(const v2f*)(up + c * O_DIM + 2 * lane);
        v2f w2 = *(const v2f*)(vp + c * O_DIM + 2 * lane);
        float p = u2.x * w2.x + u2.y * w2.y;
        for (int off = 16; off; off >>= 1) p += __shfl_xor(p, off, 32);
        if (lane == 0) b_ij[((size_t)b * R_DIM + r) * C_DIM + c] += p;
    }
}

// -------------------------------------------------------------------------
extern "C" void kernel_launch(void* const* d_in, const int* in_sizes, int n_in,
                              void* d_out, int out_size, void* d_ws, size_t ws_size,
                              hipStream_t stream) {
    const float* x = (const float*)d_in[0];   // [B,R,I]
    const float* W = (const float*)d_in[1];   // [R,C,O,I]
    float* out = (float*)d_out;               // [B,C,O,1] = 65536 f32

    // Workspace layout
    float* u_hat = (float*)d_ws;                                   // 33,554,432 f32
    float* b_ij  = u_hat + (size_t)B_DIM * R_DIM * N_DIM;          //    524,288 f32
    float* c_ij  = b_ij + (size_t)B_DIM * R_DIM * C_DIM;           //    524,288 f32
    float* s     = c_ij + (size_t)B_DIM * R_DIM * C_DIM;           //     65,536 f32
    float* vbuf  = s + (size_t)B_DIM * C_DIM * O_DIM;              //     65,536 f32

    hipMemsetAsync(b_ij, 0, (size_t)B_DIM * R_DIM * C_DIM * sizeof(float), stream);

    uhat_gemm<<<dim3(16, R_DIM), 256, 0, stream>>>(x, W, u_hat);

    for (int it = 0; it < 3; ++it) {
        softmax_c<<<dim3((B_DIM * R_DIM) / 8), 256, 0, stream>>>(b_ij, c_ij);
        reduce_s<<<dim3(C_DIM, B_DIM), 256, 0, stream>>>(c_ij, u_hat, s);
        float* vdst = (it == 2) ? out : vbuf;
        squash_v<<<dim3((B_DIM * C_DIM * O_DIM) / 256), 256, 0, stream>>>(s, vdst);
        if (it < 2) {
            agree_update<<<dim3(R_DIM, B_DIM), 256, 0, stream>>>(u_hat, vbuf, b_ij);
        }
    }
}